// GraphAttentionLayer_15341623181907
// MI455X (gfx1250) — compile-verified
//
#include <hip/hip_runtime.h>
#include <hip/hip_bf16.h>

// Problem constants (match reference)
#define NN      50000
#define IN_F    64
#define EE      800000
#define HH      4
#define FF      16
#define NEG_SLOPE 0.2f
#define EPS_GAT 1e-10f

typedef float v2f __attribute__((ext_vector_type(2)));
typedef float v8f __attribute__((ext_vector_type(8)));

// ---------------------------------------------------------------------------
// Kernel 1: h = x @ W^T using fp32 WMMA 16x16x4.
// Grid: 3125 blocks (one 16-row tile each) x 128 threads (4 waves, one per
// 16-col tile). N = 3125*16 exactly, K = 64 = 16*4 exactly -> no bounds code,
// EXEC is all-ones for every WMMA as required.
// A (16x4 f32): lanes 0-15 -> M=lane, K = k0..k0+1 ; lanes 16-31 -> K = k0+2..k0+3
// B (4x16 f32): lanes 0-15 -> K rows k0,k0+1 ; lanes 16-31 -> k0+2,k0+3 ; N = lane&15
// C/D (16x16 f32): VGPR v -> M = v + (lane>=16 ? 8 : 0), N = lane&15
// ---------------------------------------------------------------------------
__global__ void __launch_bounds__(128)
gat_gemm_wmma(const float* __restrict__ x, const float* __restrict__ W,
              float* __restrict__ h)
{
    const int rowTile = blockIdx.x;              // 0..3124
    const int wave    = threadIdx.x >> 5;        // col tile 0..3
    const int lane    = threadIdx.x & 31;
    const int m       = lane & 15;               // row in tile (A) / col in tile (B)
    const int hi      = lane >> 4;               // 0: K pair {0,1}, 1: K pair {2,3}

    const float* __restrict__ arow = x + (rowTile * 16 + m) * IN_F + hi * 2;
    const float* __restrict__ brow = W + (wave    * 16 + m) * IN_F + hi * 2;

    v8f acc = {};
#pragma unroll
    for (int k0 = 0; k0 < IN_F; k0 += 4) {
        v2f a = *(const v2f*)(arow + k0);        // x[row, k0+2*hi .. +1]
        v2f b = *(const v2f*)(brow + k0);        // W[col, k0+2*hi .. +1]  (== B^T slice)
        acc = __builtin_amdgcn_wmma_f32_16x16x4_f32(
                  /*neg_a=*/false, a, /*neg_b=*/false, b,
                  /*c_mod=*/(short)0, acc, /*reuse_a=*/false, /*reuse_b=*/false);
    }

    const int n     = lane & 15;
    const int mBase = rowTile * 16 + hi * 8;
    float* __restrict__ dst = h + (size_t)mBase * IN_F + wave * 16 + n;
#pragma unroll
    for (int v = 0; v < 8; ++v)
        dst[(size_t)v * IN_F] = acc[v];
}

// ---------------------------------------------------------------------------
// Kernel 2: per-node logits  as[n,h] = <h[n,h,:], a_src[h,:]>, same for a_dst.
// One thread per (node, head): 64B contiguous read, 2 dot16.
// ---------------------------------------------------------------------------
__global__ void gat_node_logits(const float* __restrict__ h,
                                const float* __restrict__ a_src,
                                const float* __restrict__ a_dst,
                                float* __restrict__ asn,
                                float* __restrict__ adn)
{
    int tid = blockIdx.x * blockDim.x + threadIdx.x;
    if (tid >= NN * HH) return;
    const int head = tid & (HH - 1);
    const float4* __restrict__ hp = (const float4*)(h + (size_t)tid * FF);
    const float4* __restrict__ sp = (const float4*)(a_src + head * FF);
    const float4* __restrict__ dp = (const float4*)(a_dst + head * FF);
    float s = 0.f, d = 0.f;
#pragma unroll
    for (int q = 0; q < 4; ++q) {
        float4 hv = hp[q], sv = sp[q], dv = dp[q];
        s += hv.x * sv.x + hv.y * sv.y + hv.z * sv.z + hv.w * sv.w;
        d += hv.x * dv.x + hv.y * dv.y + hv.z * dv.z + hv.w * dv.w;
    }
    asn[tid] = s;
    adn[tid] = d;
}

// Order-preserving float<->uint map so unsigned atomicMax == float max.
__device__ __forceinline__ unsigned fmax_enc(float f) {
    unsigned u = __float_as_uint(f);
    return (u & 0x80000000u) ? ~u : (u | 0x80000000u);
}
__device__ __forceinline__ float fmax_dec(unsigned k) {
    unsigned u = (k & 0x80000000u) ? (k ^ 0x80000000u) : ~k;
    return __uint_as_float(u);
}

// ---------------------------------------------------------------------------
// Kernel 3: per-(edge,head) alpha = leaky_relu(as[src]+ad[dst]); segment max.
// ---------------------------------------------------------------------------
__global__ void gat_edge_alpha_max(const long long* __restrict__ ei,
                                   const float* __restrict__ asn,
                                   const float* __restrict__ adn,
                                   float* __restrict__ ebuf,
                                   unsigned* __restrict__ segmax)
{
    int tid = blockIdx.x * blockDim.x + threadIdx.x;
    if (tid >= EE * HH) return;
    const int e = tid >> 2, head = tid & 3;
    const long long s = ei[e];
    const long long d = ei[EE + e];
    float a = asn[s * HH + head] + adn[d * HH + head];
    float alpha = (a > 0.f) ? a : a * NEG_SLOPE;
    ebuf[tid] = alpha;
    atomicMax(&segmax[d * HH + head], fmax_enc(alpha));
}

// ---------------------------------------------------------------------------
// Kernel 4: ex = exp(alpha - segmax[dst]); segment sum. ebuf: alpha -> ex.
// ---------------------------------------------------------------------------
__global__ void gat_edge_exp_sum(const long long* __restrict__ ei,
                                 const unsigned* __restrict__ segmax,
                                 float* __restrict__ ebuf,
                                 float* __restrict__ segsum)
{
    int tid = blockIdx.x * blockDim.x + threadIdx.x;
    if (tid >= EE * HH) return;
    const int e = tid >> 2, head = tid & 3;
    const long long d = ei[EE + e];
    float mx = fmax_dec(segmax[d * HH + head]);
    float ex = expf(ebuf[tid] - mx);
    ebuf[tid] = ex;
    atomicAdd(&segsum[d * HH + head], ex);
}

// ---------------------------------------------------------------------------
// Kernel 5: out[dst,h,:] += h[src,h,:] * ex/(segsum[dst,h]+eps)
// One thread per (edge,head): 64B h[src] gather + 16 f32 atomics.
// ---------------------------------------------------------------------------
__global__ void gat_edge_aggregate(const long long* __restrict__ ei,
                                   const float* __restrict__ h,
                                   const float* __restrict__ ebuf,
                                   const float* __restrict__ segsum,
                                   float* __restrict__ out)
{
    int tid = blockIdx.x * blockDim.x + threadIdx.x;
    if (tid >= EE * HH) return;
    const int e = tid >> 2, head = tid & 3;
    const long long s = ei[e];
    const long long d = ei[EE + e];
    float att = ebuf[tid] / (segsum[d * HH + head] + EPS_GAT);

    const float4* __restrict__ hp = (const float4*)(h + (size_t)s * IN_F + head * FF);
    float* __restrict__ op = out + (size_t)d * IN_F + head * FF;
#pragma unroll
    for (int q = 0; q < 4; ++q) {
        float4 hv = hp[q];
        atomicAdd(&op[q * 4 + 0], hv.x * att);
        atomicAdd(&op[q * 4 + 1], hv.y * att);
        atomicAdd(&op[q * 4 + 2], hv.z * att);
        atomicAdd(&op[q * 4 + 3], hv.w * att);
    }
}

// ---------------------------------------------------------------------------
extern "C" void kernel_launch(void* const* d_in, const int* in_sizes, int n_in,
                              void* d_out, int out_size, void* d_ws, size_t ws_size,
                              hipStream_t stream)
{
    const float*     x     = (const float*)d_in[0];
    const long long* ei    = (const long long*)d_in[1];   // (2,E) int64: [0..E)=src, [E..2E)=dst
    const float*     W     = (const float*)d_in[2];
    const float*     a_src = (const float*)d_in[3];
    const float*     a_dst = (const float*)d_in[4];
    float*           out   = (float*)d_out;

    // Workspace layout (all offsets multiples of 256B). Total = 28.8 MB.
    char* ws = (char*)d_ws;
    float*    hbuf   = (float*)(ws + 0);           // N*64*4      = 12,800,000
    float*    asn    = (float*)(ws + 12800000);    // N*H*4       =    800,000
    float*    adn    = (float*)(ws + 13600000);    // N*H*4       =    800,000
    unsigned* segmax = (unsigned*)(ws + 14400000); // N*H*4       =    800,000
    float*    segsum = (float*)(ws + 15200000);    // N*H*4       =    800,000
    float*    ebuf   = (float*)(ws + 16000000);    // E*H*4       = 12,800,000

    // Atomics accumulate: re-zero accumulators every call (graph-capture safe).
    hipMemsetAsync(out,    0, (size_t)NN * IN_F * sizeof(float), stream);
    hipMemsetAsync(segmax, 0, (size_t)NN * HH * sizeof(unsigned), stream); // enc: any finite > 0
    hipMemsetAsync(segsum, 0, (size_t)NN * HH * sizeof(float), stream);

    // 1) h = x @ W^T  (fp32 WMMA)
    gat_gemm_wmma<<<NN / 16, 128, 0, stream>>>(x, W, hbuf);

    // 2) per-node attention logits
    gat_node_logits<<<(NN * HH + 255) / 256, 256, 0, stream>>>(hbuf, a_src, a_dst, asn, adn);

    // 3) per-edge alpha + segment max
    gat_edge_alpha_max<<<(EE * HH + 255) / 256, 256, 0, stream>>>(ei, asn, adn, ebuf, segmax);

    // 4) per-edge exp + segment sum
    gat_edge_exp_sum<<<(EE * HH + 255) / 256, 256, 0, stream>>>(ei, segmax, ebuf, segsum);

    // 5) weighted scatter-add into out
    gat_edge_aggregate<<<(EE * HH + 255) / 256, 256, 0, stream>>>(ei, hbuf, ebuf, segsum, out);
}